// StaticPhaseDifferenceComputer_85907935855224
// MI455X (gfx1250) — compile-verified
//
#include <hip/hip_runtime.h>
#include <math.h>

typedef __attribute__((ext_vector_type(2))) float v2f;
typedef __attribute__((ext_vector_type(4))) float v4f;
typedef __attribute__((ext_vector_type(8))) float v8f;

#define B_   32
#define S_   2048
#define N_   16
#define D_   128
#define ND_  (N_ * D_)      // 2048 floats per (b,s) slab
#define H_   64
#define NC_  16             // S-chunks per batch
#define SC_  (S_ / NC_)     // 128 s-values per chunk

// ---------------------------------------------------------------------------
// Kernel 1: partial sums over S.  grid = (NC_, B_), block = 512 threads.
// Each block owns one (batch, s-chunk); thread t owns float4 column t of the
// 2048-wide [N*D] row and accumulates SC_ rows.  Fully coalesced B128 loads,
// non-temporal (streamed once, 536 MB total -> this IS the roofline).
// ---------------------------------------------------------------------------
__global__ __launch_bounds__(512) void mean_psum_kernel(
    const float* __restrict__ x, float* __restrict__ part) {
  const int c = blockIdx.x;   // chunk
  const int b = blockIdx.y;   // batch
  const int t = threadIdx.x;  // 0..511 -> float4 column

  const v4f* xp = (const v4f*)(x + ((size_t)b * S_ + (size_t)c * SC_) * ND_) + t;
  v4f acc = {0.f, 0.f, 0.f, 0.f};
  #pragma unroll 8
  for (int s = 0; s < SC_; ++s) {
    v4f v = __builtin_nontemporal_load(xp);
    acc += v;
    xp += ND_ / 4;
  }
  v4f* pp = (v4f*)(part + ((size_t)b * NC_ + c) * ND_) + t;
  *pp = acc;  // regular store: stays warm in L2 for kernel 2
}

// ---------------------------------------------------------------------------
// Kernel 2: per-batch.  Reduce partials -> mean in LDS, then wave 0 computes
// P = M*W1a and Q = M*W1b (16x128 @ 128x64) with V_WMMA_F32_16X16X4_F32,
// then all 256 threads evaluate one edge (i,j) each.
//
// WMMA layouts used (CDNA5 ISA 7.12.2, wave32):
//   A 16x4 f32 : lane L -> M = L%16 ; K = (L/16)*2 + vgpr(0..1)
//   B  4x16 f32: lane L -> N = L%16 ; K = (L/16)*2 + vgpr(0..1)
//   C 16x16 f32: lane L -> N = L%16 ; M = (L/16)*8 + vgpr(0..7)
// ---------------------------------------------------------------------------
__global__ __launch_bounds__(256) void edge_mlp_kernel(
    const float* __restrict__ part,
    const float* __restrict__ W1, const float* __restrict__ b1,
    const float* __restrict__ W2, const float* __restrict__ b2,
    float* __restrict__ out) {
  __shared__ float means[ND_];   // 8 KB  : mean matrix M [16][128]
  __shared__ float Ps[N_ * H_];  // 4 KB  : P [16][64]
  __shared__ float Qs[N_ * H_];  // 4 KB  : Q [16][64]
  __shared__ float b1s[H_];
  __shared__ float w2s[H_];
  __shared__ float b2s;

  const int b   = blockIdx.x;
  const int tid = threadIdx.x;

  if (tid < H_) { b1s[tid] = b1[tid]; w2s[tid] = W2[tid]; }
  if (tid == 0) { b2s = b2[0]; }

  // ---- reduce 16 partials -> mean (512 float4, 2 per thread) ----
  const v4f* pv = (const v4f*)(part + (size_t)b * NC_ * ND_);
  v4f* mv = (v4f*)means;
  const float inv_s = 1.0f / (float)S_;
  #pragma unroll
  for (int g = 0; g < 2; ++g) {
    const int i4 = g * 256 + tid;
    v4f acc = {0.f, 0.f, 0.f, 0.f};
    #pragma unroll
    for (int c = 0; c < NC_; ++c) acc += pv[c * (ND_ / 4) + i4];
    acc *= inv_s;
    mv[i4] = acc;
  }
  __syncthreads();

  // ---- wave 0: P and Q via fp32 WMMA ----
  if (tid < 32) {
    const int lane = tid;
    const int nn = lane & 15;   // M for A-frag, N for B/C-frags
    const int kh = lane >> 4;   // K-half select

    #pragma unroll
    for (int half = 0; half < 2; ++half) {
      const int base = half * D_;           // W1a rows 0..127 / W1b rows 128..255
      float* dst = half ? Qs : Ps;
      v8f c0 = {}, c1 = {}, c2 = {}, c3 = {};

      for (int kb = 0; kb < D_; kb += 4) {
        const int k0 = kb + kh * 2;         // this lane's first K
        v2f a;
        a.x = means[nn * D_ + k0];
        a.y = means[nn * D_ + k0 + 1];
        const float* w1r0 = W1 + (size_t)(base + k0) * H_;       // row k0
        const float* w1r1 = w1r0 + H_;                           // row k0+1

        v2f bt;
        bt.x = w1r0[0 * 16 + nn]; bt.y = w1r1[0 * 16 + nn];
        c0 = __builtin_amdgcn_wmma_f32_16x16x4_f32(false, a, false, bt,
                                                   (short)0, c0, false, false);
        bt.x = w1r0[1 * 16 + nn]; bt.y = w1r1[1 * 16 + nn];
        c1 = __builtin_amdgcn_wmma_f32_16x16x4_f32(false, a, false, bt,
                                                   (short)0, c1, false, false);
        bt.x = w1r0[2 * 16 + nn]; bt.y = w1r1[2 * 16 + nn];
        c2 = __builtin_amdgcn_wmma_f32_16x16x4_f32(false, a, false, bt,
                                                   (short)0, c2, false, false);
        bt.x = w1r0[3 * 16 + nn]; bt.y = w1r1[3 * 16 + nn];
        c3 = __builtin_amdgcn_wmma_f32_16x16x4_f32(false, a, false, bt,
                                                   (short)0, c3, false, false);
      }

      // scatter C tiles to LDS: lane L, vgpr v -> (m = kh*8+v, n = nn)
      #pragma unroll
      for (int v = 0; v < 8; ++v) {
        const int m = kh * 8 + v;
        dst[m * H_ +  0 + nn] = c0[v];
        dst[m * H_ + 16 + nn] = c1[v];
        dst[m * H_ + 32 + nn] = c2[v];
        dst[m * H_ + 48 + nn] = c3[v];
      }
    }
  }
  __syncthreads();

  // ---- edge epilogue: thread tid handles pair (i = tid/16, j = tid%16) ----
  const int i = tid >> 4;
  const int j = tid & 15;
  float* ob = out + (size_t)b * N_ * N_;
  if (i == j) {
    ob[i * N_ + j] = 0.0f;                        // diagonal stays 0
  } else if (i < j) {
    float acc = 0.0f;
    #pragma unroll 8
    for (int k = 0; k < H_; ++k) {
      const float hpre = Ps[i * H_ + k] + Qs[j * H_ + k] + b1s[k];
      // exact-erf GELU (torch / jax approximate=False)
      const float g = 0.5f * hpre * (1.0f + erff(hpre * 0.70710678118654752f));
      acc += g * w2s[k];
    }
    const float z = acc + b2s;
    const float s = 1.0f / (1.0f + expf(-z));
    ob[i * N_ + j] = s;                           // upper
    ob[j * N_ + i] = s;                           // mirrored lower
  }
  // i > j: written by the (j,i) thread above
}

// ---------------------------------------------------------------------------
extern "C" void kernel_launch(void* const* d_in, const int* in_sizes, int n_in,
                              void* d_out, int out_size, void* d_ws, size_t ws_size,
                              hipStream_t stream) {
  const float* x  = (const float*)d_in[0];  // [32,2048,16,128]
  const float* W1 = (const float*)d_in[1];  // [256,64]
  const float* b1 = (const float*)d_in[2];  // [64]
  const float* W2 = (const float*)d_in[3];  // [64,1]
  const float* b2 = (const float*)d_in[4];  // [1]
  float* out  = (float*)d_out;              // [32,16,16]
  float* part = (float*)d_ws;               // [32,16,2048] = 4 MB partial sums

  mean_psum_kernel<<<dim3(NC_, B_), 512, 0, stream>>>(x, part);
  edge_mlp_kernel<<<B_, 256, 0, stream>>>(part, W1, b1, W2, b2, out);
}